// DenoiseNet_77592879169623
// MI455X (gfx1250) — compile-verified
//
#include <hip/hip_runtime.h>

typedef float v2f __attribute__((ext_vector_type(2)));
typedef float v8f __attribute__((ext_vector_type(8)));

#define HDIM 64
#define WPAD 66                  /* LDS row pad: even (8B-aligned pairs) + bank spread */
#define GAMMA_C (-0.45f)
#define ZG_C    (1.5f)           /* ZETA - GAMMA = 1.05 - (-0.45) */

__global__ void zero_kernel(float* __restrict__ p, int n) {
    int i = blockIdx.x * blockDim.x + threadIdx.x;
    int stride = gridDim.x * blockDim.x;
    for (; i < n; i += stride) p[i] = 0.0f;
}

// Per-node attention scores via WMMA f32 16x16x4.
// s1[n] = relu(x[n] @ W_nb   + b_nb  ) . Watt[0:64]
// s2[n] = relu(x[n] @ W_self + b_self) . Watt[64:128]
// Block stages both weight matrices (transposed, padded) into LDS once; each
// wave grid-strides over 16-node tiles. Requires N % 16 == 0 (100000 = 16*6250).
__global__ __launch_bounds__(128) void score_kernel(
    const float* __restrict__ x,
    const float* __restrict__ Wnb,   const float* __restrict__ bnb,
    const float* __restrict__ Wself, const float* __restrict__ bself,
    const float* __restrict__ Watt,   // [128]
    float* __restrict__ s1, float* __restrict__ s2, int n_tiles)
{
    __shared__ float lW[2][64 * WPAD];   // lW[p][c*WPAD + k] = W_p[k*64 + c]  (transposed)
    __shared__ float lb[2][64];
    __shared__ float la[2][64];

    const int tid = threadIdx.x;
    for (int idx = tid; idx < 64 * 64; idx += 128) {
        const int k = idx >> 6, c = idx & 63;
        lW[0][c * WPAD + k] = Wnb[idx];
        lW[1][c * WPAD + k] = Wself[idx];
    }
    if (tid < 64) {
        lb[0][tid] = bnb[tid];
        lb[1][tid] = bself[tid];
        la[0][tid] = Watt[tid];
        la[1][tid] = Watt[64 + tid];
    }
    __syncthreads();

    const int lane = tid & 31;
    const int wave = tid >> 5;
    const int m  = lane & 15;            // A: M row / B: N col / bias+att index
    const int kh = lane >> 4;            // 0 -> K pair {0,1}; 1 -> {2,3} of each K4 chunk
    const int wstride = gridDim.x * 4;

    for (int tile = blockIdx.x * 4 + wave; tile < n_tiles; tile += wstride) {
        const int node0 = tile * 16;

        // A fragments (ISA 32-bit A 16x4 layout), 16 chunks cover K=0..63
        v2f afr[16];
        const float* ap = x + (size_t)(node0 + m) * HDIM + 2 * kh;
        #pragma unroll
        for (int t = 0; t < 16; ++t) afr[t] = *(const v2f*)(ap + 4 * t);

        #pragma unroll
        for (int pass = 0; pass < 2; ++pass) {
            float* sout = pass ? s2 : s1;
            float acc[8];
            #pragma unroll
            for (int r = 0; r < 8; ++r) acc[r] = 0.0f;

            #pragma unroll
            for (int j = 0; j < 4; ++j) {            // output cols j*16 .. j*16+15
                const int c0 = (j * 16 + m) * WPAD;  // transposed weight row in LDS
                v8f c = {};
                #pragma unroll
                for (int t = 0; t < 16; ++t) {       // K chunks of 4
                    v2f bfr = *(const v2f*)&lW[pass][c0 + 4 * t + 2 * kh];
                    c = __builtin_amdgcn_wmma_f32_16x16x4_f32(
                            false, afr[t], false, bfr, (short)0, c, false, false);
                }
                // C/D layout: lanes 0-15 vgpr r -> (M=r, N=lane); lanes 16-31 -> (M=r+8, N=lane-16)
                const float bias = lb[pass][j * 16 + m];
                const float av   = la[pass][j * 16 + m];
                #pragma unroll
                for (int r = 0; r < 8; ++r) {
                    float u = c[r] + bias;
                    u = u > 0.0f ? u : 0.0f;         // relu
                    acc[r] = fmaf(u, av, acc[r]);    // dot with attention segment
                }
            }
            // reduce across the 16 lanes of each half
            #pragma unroll
            for (int r = 0; r < 8; ++r) {
                float v = acc[r];
                v += __shfl_xor(v, 1, 32);
                v += __shfl_xor(v, 2, 32);
                v += __shfl_xor(v, 4, 32);
                v += __shfl_xor(v, 8, 32);
                acc[r] = v;
            }
            if (m == 0) {
                const int base = node0 + kh * 8;
                #pragma unroll
                for (int r = 0; r < 8; ++r) sout[base + r] = acc[r];
            }
        }
    }
}

__global__ void edge_gate_kernel(
    const long long* __restrict__ row, const long long* __restrict__ col,
    const float* __restrict__ s1, const float* __restrict__ s2,
    const float* __restrict__ batt,
    float* __restrict__ maskb, float* __restrict__ rowsum, int nE)
{
    int e = blockIdx.x * blockDim.x + threadIdx.x;
    if (e >= nE) return;
    int r = (int)row[e], c = (int)col[e];
    float logit = s1[r] + s2[c] + batt[0];
    float gate  = 1.0f / (1.0f + expf(-logit));
    float mk    = fminf(fmaxf(gate * ZG_C + GAMMA_C, 0.0f), 1.0f);
    maskb[e] = mk;
    atomicAdd(rowsum + r, mk);
}

__global__ void dinv_kernel(const float* __restrict__ rowsum, float* __restrict__ dinv, int n) {
    int i = blockIdx.x * blockDim.x + threadIdx.x;
    if (i >= n) return;
    float d = rsqrtf(rowsum[i] + 1e-6f);
    dinv[i] = fminf(d, 10.0f);
}

// out[row[e]] += mask[e]*dinv[row]*dinv[col] * xin[col[e]] ; 16 threads/edge, float4 each
__global__ void spmm_kernel(
    const long long* __restrict__ row, const long long* __restrict__ col,
    const float* __restrict__ maskb, const float* __restrict__ dinv,
    const float* __restrict__ xin, float* __restrict__ xout, int nE)
{
    int tid = blockIdx.x * blockDim.x + threadIdx.x;
    int e = tid >> 4;
    int t = tid & 15;
    if (e >= nE) return;
    int r = (int)row[e], c = (int)col[e];
    float v = maskb[e] * dinv[r] * dinv[c];
    const float4 xv = *(const float4*)(xin + (size_t)c * HDIM + t * 4);
    float* op = xout + (size_t)r * HDIM + t * 4;
    atomicAdd(op + 0, v * xv.x);
    atomicAdd(op + 1, v * xv.y);
    atomicAdd(op + 2, v * xv.z);
    atomicAdd(op + 3, v * xv.w);
}

__global__ void combine_kernel(const float* __restrict__ x0, const float* __restrict__ x1,
                               const float* __restrict__ x2, float* __restrict__ out, int n) {
    int i = blockIdx.x * blockDim.x + threadIdx.x;
    if (i < n) out[i] = x0[i] + x1[i] + x2[i];
}

extern "C" void kernel_launch(void* const* d_in, const int* in_sizes, int n_in,
                              void* d_out, int out_size, void* d_ws, size_t ws_size,
                              hipStream_t stream)
{
    const float*     x0  = (const float*)d_in[0];
    const long long* row = (const long long*)d_in[1];   // int64 indices
    const long long* col = (const long long*)d_in[2];
    const float* Wnb0   = (const float*)d_in[3];
    const float* bnb0   = (const float*)d_in[4];
    const float* Wself0 = (const float*)d_in[5];
    const float* bself0 = (const float*)d_in[6];
    const float* Watt0  = (const float*)d_in[7];
    const float* batt0  = (const float*)d_in[8];
    const float* Wnb1   = (const float*)d_in[9];
    const float* bnb1   = (const float*)d_in[10];
    const float* Wself1 = (const float*)d_in[11];
    const float* bself1 = (const float*)d_in[12];
    const float* Watt1  = (const float*)d_in[13];
    const float* batt1  = (const float*)d_in[14];

    const int N  = in_sizes[0] / HDIM;   // 100000
    const int nE = in_sizes[1];          // 800000
    const int NT = N / 16;               // node tiles (N % 16 == 0 here)

    // workspace layout (floats)
    float* ws     = (float*)d_ws;
    float* x1     = ws;
    float* x2     = x1 + (size_t)N * HDIM;
    float* s1v    = x2 + (size_t)N * HDIM;
    float* s2v    = s1v + N;
    float* rowsum = s2v + N;
    float* dinv   = rowsum + N;
    float* maskb  = dinv + N;

    const dim3 b256(256);
    const dim3 b128(128);
    int gScore = (NT + 3) / 4;
    if (gScore > 640) gScore = 640;      // grid-stride tiles; amortizes LDS weight staging
    const int gEdge = (nE + 255) / 256;
    const int gNode = (N + 255) / 256;
    const int gSpmm = (nE * 16 + 255) / 256;
    const int gElem = (N * HDIM + 255) / 256;

    // ---- layer 0 : x0 -> x1 ----
    zero_kernel<<<2048, b256, 0, stream>>>(x1, N * HDIM);
    zero_kernel<<<256,  b256, 0, stream>>>(rowsum, N);
    score_kernel<<<gScore, b128, 0, stream>>>(x0, Wnb0, bnb0, Wself0, bself0, Watt0, s1v, s2v, NT);
    edge_gate_kernel<<<gEdge, b256, 0, stream>>>(row, col, s1v, s2v, batt0, maskb, rowsum, nE);
    dinv_kernel<<<gNode, b256, 0, stream>>>(rowsum, dinv, N);
    spmm_kernel<<<gSpmm, b256, 0, stream>>>(row, col, maskb, dinv, x0, x1, nE);

    // ---- layer 1 : x1 -> x2 ----
    zero_kernel<<<2048, b256, 0, stream>>>(x2, N * HDIM);
    zero_kernel<<<256,  b256, 0, stream>>>(rowsum, N);
    score_kernel<<<gScore, b128, 0, stream>>>(x1, Wnb1, bnb1, Wself1, bself1, Watt1, s1v, s2v, NT);
    edge_gate_kernel<<<gEdge, b256, 0, stream>>>(row, col, s1v, s2v, batt1, maskb, rowsum, nE);
    dinv_kernel<<<gNode, b256, 0, stream>>>(rowsum, dinv, N);
    spmm_kernel<<<gSpmm, b256, 0, stream>>>(row, col, maskb, dinv, x1, x2, nE);

    // ---- out = x0 + x1 + x2 ----
    combine_kernel<<<gElem, b256, 0, stream>>>(x0, x1, x2, (float*)d_out, N * HDIM);
}